// GCFNN_8753143349492
// MI455X (gfx1250) — compile-verified
//
#include <hip/hip_runtime.h>
#include <hip/hip_bf16.h>

typedef float v2f __attribute__((ext_vector_type(2)));
typedef float v8f __attribute__((ext_vector_type(8)));
typedef unsigned int u32x4 __attribute__((ext_vector_type(4)));
typedef unsigned int u32x8 __attribute__((ext_vector_type(8)));

#define NN 4096
#define DD 400
#define HH 256
#define FF 128
#define ZZ 64
#define YY 5
#define PHH 128

__device__ __forceinline__ float lrelu(float x) { return x >= 0.0f ? x : 0.25f * x; }

// ---- CDNA5 async global->LDS copy (ASYNCcnt path), 16B per lane -------------
__device__ __forceinline__ void async_ld16(void* lds, const void* g) {
    unsigned l = (unsigned)(uintptr_t)lds;  // low 32 bits of generic addr = LDS offset
    asm volatile("global_load_async_to_lds_b128 %0, %1, off"
                 :: "v"(l), "v"(g) : "memory");
}
__device__ __forceinline__ void async_wait0() {
    asm volatile("s_wait_asynccnt 0" ::: "memory");
}

// ---- CDNA5 Tensor Data Mover: 2D tile global->LDS (TENSORcnt path) ---------
// D# per CDNA5 ISA. 4-byte elements.
// tile0 x tile1 tile from a [dim1 x dim0] row-major tensor, row stride = stride0
// elements; LDS rows padded: padInt/padAmt are the encoded pad codes.
__device__ __forceinline__ void tdm_load_2d(
    unsigned ldsOff, const void* gaddr,
    unsigned dim0, unsigned dim1,
    unsigned tile0, unsigned tile1, unsigned stride0,
    unsigned padInt, unsigned padAmt)
{
    unsigned long long ga = (unsigned long long)(uintptr_t)gaddr;
    u32x4 g0;
    g0[0] = 1u;                                  // count=1, user descriptor
    g0[1] = ldsOff;                              // lds_addr (bytes)
    g0[2] = (unsigned)ga;                        // global_addr[31:0]
    g0[3] = (unsigned)(ga >> 32) | (2u << 30);   // global_addr[56:32] | type=2
    u32x8 g1;
    g1[0] = (2u << 16) | (1u << 20) | (padInt << 22) | (padAmt << 25); // 4B, pad
    g1[1] = (dim0 & 0xFFFFu) << 16;                       // tensor_dim0 lo
    g1[2] = (dim0 >> 16) | ((dim1 & 0xFFFFu) << 16);      // dim0 hi | dim1 lo
    g1[3] = (dim1 >> 16) | (tile0 << 16);                 // dim1 hi | tile_dim0
    g1[4] = tile1;                                        // tile_dim1 (tile_dim2=0)
    g1[5] = stride0;                                      // tensor_dim0_stride lo
    g1[6] = 0;
    g1[7] = 0;
    asm volatile("tensor_load_to_lds %0, %1" :: "s"(g0), "s"(g1) : "memory");
}

// ---------------------------------------------------------------------------
// fp32 WMMA GEMM:  C = act(A[MxK] @ B[KxN] + bias[N])
// BM=128, BN=128, BK=16. 256 threads = 8 waves (4 along M x 2 along N);
// each wave computes 32x64 = 2x4 v_wmma_f32_16x16x4_f32 accumulators.
// Double-buffered LDS: A tiles streamed by the TDM (wave 0 issues, TENSORcnt),
// B tiles prefetched to registers and scatter-stored transposed (Blds[n][k],
// stride 20) so every fragment is one aligned ds_load_b64.
// Requires M%128==0, N%128==0, K%16==0 (true for all calls here).
// ---------------------------------------------------------------------------
__global__ __launch_bounds__(256) void gemm_wmma_f32(
    const float* __restrict__ A, const float* __restrict__ B,
    const float* __restrict__ bias, float* __restrict__ C,
    int M, int N, int K, int hasBias, int act)
{
    __shared__ float Alds[2][128 * 20]; // [m][k] rows padded to 20 by TDM
    __shared__ float Blds[2][128 * 20]; // [n][k] transposed

    const int tid  = threadIdx.x;
    const int lane = tid & 31;
    const int wave = tid >> 5;
    const int wm   = wave & 3;   // wave row -> *32
    const int wn   = wave >> 2;  // wave col -> *64
    const int half = lane >> 4;
    const int l16  = lane & 15;
    const bool w0  = (wave == 0);

    const int m0 = blockIdx.y * 128;
    const int n0 = blockIdx.x * 128;

    const int nIter = K / 16;

    v8f acc[2][4];
#pragma unroll
    for (int i = 0; i < 2; ++i)
#pragma unroll
        for (int j = 0; j < 4; ++j) acc[i][j] = (v8f)(0.0f);

    // B-tile register prefetch state (16x128 tile -> 2 float4 per thread)
    const int br0 = (tid * 2) >> 5, bc0 = (tid * 2) & 31;
    const int br1 = (tid * 2 + 1) >> 5, bc1 = (tid * 2 + 1) & 31;
    float4 breg0, breg1;

    // ---- prologue: stage chunk 0 into buffer 0
    breg0 = *(const float4*)(B + (size_t)br0 * N + n0 + bc0 * 4);
    breg1 = *(const float4*)(B + (size_t)br1 * N + n0 + bc1 * 4);
    if (w0)
        tdm_load_2d((unsigned)(uintptr_t)&Alds[0][0],
                    A + (size_t)m0 * K, (unsigned)K, (unsigned)M,
                    16u, 128u, (unsigned)K, 3u, 3u);     // 16 dwords + 4 pad
    {
        float* bl = &Blds[0][0];
        bl[(bc0 * 4 + 0) * 20 + br0] = breg0.x;
        bl[(bc0 * 4 + 1) * 20 + br0] = breg0.y;
        bl[(bc0 * 4 + 2) * 20 + br0] = breg0.z;
        bl[(bc0 * 4 + 3) * 20 + br0] = breg0.w;
        bl[(bc1 * 4 + 0) * 20 + br1] = breg1.x;
        bl[(bc1 * 4 + 1) * 20 + br1] = breg1.y;
        bl[(bc1 * 4 + 2) * 20 + br1] = breg1.z;
        bl[(bc1 * 4 + 3) * 20 + br1] = breg1.w;
    }
    if (w0) __builtin_amdgcn_s_wait_tensorcnt(0);
    __syncthreads();

    for (int i = 0; i < nIter; ++i) {
        const int cur = i & 1;
        const int nxt = cur ^ 1;
        const bool more = (i + 1 < nIter);

        // ---- issue next chunk's fetches (overlap with this chunk's WMMAs)
        if (more) {
            const int k1 = (i + 1) * 16;
            breg0 = *(const float4*)(B + (size_t)(k1 + br0) * N + n0 + bc0 * 4);
            breg1 = *(const float4*)(B + (size_t)(k1 + br1) * N + n0 + bc1 * 4);
            if (w0)
                tdm_load_2d((unsigned)(uintptr_t)&Alds[nxt][0],
                            A + (size_t)m0 * K + k1, (unsigned)K, (unsigned)M,
                            16u, 128u, (unsigned)K, 3u, 3u);
        }

        // ---- compute 128x128 += 128x16 @ 16x128 (32 WMMAs/wave)
        const float* Al = &Alds[cur][0];
        const float* Bl = &Blds[cur][0];
#pragma unroll
        for (int ks = 0; ks < 16; ks += 4) {
            v2f a0, a1, b[4];
            const float* ap0 = &Al[(wm * 32 + l16) * 20 + ks + 2 * half];
            a0.x = ap0[0]; a0.y = ap0[1];
            const float* ap1 = ap0 + 16 * 20;
            a1.x = ap1[0]; a1.y = ap1[1];
#pragma unroll
            for (int t = 0; t < 4; ++t) {
                const float* bp = &Bl[(wn * 64 + t * 16 + l16) * 20 + ks + 2 * half];
                b[t].x = bp[0]; b[t].y = bp[1];
            }
#pragma unroll
            for (int t = 0; t < 4; ++t) {
                acc[0][t] = __builtin_amdgcn_wmma_f32_16x16x4_f32(false, a0, false, b[t], (short)0, acc[0][t], false, false);
                acc[1][t] = __builtin_amdgcn_wmma_f32_16x16x4_f32(false, a1, false, b[t], (short)0, acc[1][t], false, false);
            }
        }

        // ---- finish staging next buffer, then one barrier
        if (more) {
            float* bl = &Blds[nxt][0];
            bl[(bc0 * 4 + 0) * 20 + br0] = breg0.x;
            bl[(bc0 * 4 + 1) * 20 + br0] = breg0.y;
            bl[(bc0 * 4 + 2) * 20 + br0] = breg0.z;
            bl[(bc0 * 4 + 3) * 20 + br0] = breg0.w;
            bl[(bc1 * 4 + 0) * 20 + br1] = breg1.x;
            bl[(bc1 * 4 + 1) * 20 + br1] = breg1.y;
            bl[(bc1 * 4 + 2) * 20 + br1] = breg1.z;
            bl[(bc1 * 4 + 3) * 20 + br1] = breg1.w;
            if (w0) __builtin_amdgcn_s_wait_tensorcnt(0);
        }
        __syncthreads();
    }

    // ---- epilogue: D layout: vgpr v, half h, col l16 -> row = v + 8*h
#pragma unroll
    for (int ti = 0; ti < 2; ++ti) {
#pragma unroll
        for (int t = 0; t < 4; ++t) {
            int col = n0 + wn * 64 + t * 16 + l16;
            float bv = hasBias ? bias[col] : 0.0f;
#pragma unroll
            for (int v = 0; v < 8; ++v) {
                int row = m0 + wm * 32 + ti * 16 + v + 8 * half;
                float x = acc[ti][t][v] + bv;
                if (act) x = lrelu(x);
                C[(size_t)row * N + col] = x;
            }
        }
    }
}

// ---------------------------------------------------------------------------
// s1[i] = H[i,:] . a[0:128] ; s2[i] = H[i,:] . a[128:256]
// ---------------------------------------------------------------------------
__global__ __launch_bounds__(256) void rowdots(
    const float* __restrict__ Hb, const float* __restrict__ a,
    float* __restrict__ s1, float* __restrict__ s2)
{
    int i = blockIdx.x * 256 + threadIdx.x;
    const float* h = Hb + (size_t)i * FF;
    float a1 = 0.0f, a2 = 0.0f;
#pragma unroll 8
    for (int f = 0; f < FF; ++f) {
        float v = h[f];
        a1 = fmaf(v, a[f], a1);
        a2 = fmaf(v, a[FF + f], a2);
    }
    s1[i] = a1;
    s2[i] = a2;
}

// ---------------------------------------------------------------------------
// GAT attention, flash-style streaming softmax.
// Block = 256 threads = 32 rows x 8 threads; 64-column chunks.
// e = lrelu(s1[i]+s2[j]); adj<=0 -> -1e12; online softmax; acc += P@H;
// out = lrelu(acc / l).  Tile staging via per-lane async-to-LDS.
// ---------------------------------------------------------------------------
__global__ __launch_bounds__(256) void gat_attention(
    const float* __restrict__ adj, const float* __restrict__ Hb,
    const float* __restrict__ s1, const float* __restrict__ s2,
    float* __restrict__ Out)
{
    __shared__ float s2c[64];
    __shared__ float Hc[64 * FF];   // 32 KB
    __shared__ float Pl[32 * 64];   // adj tile, then P tile (same per-slot owner)

    const int tid = threadIdx.x;
    const int row = tid >> 3;   // 0..31
    const int g   = tid & 7;    // 0..7
    const int f0  = g * 16;
    const int r0  = blockIdx.x * 32;

    const float s1v = s1[r0 + row];

    float mrun = -3.0e38f;
    float lrun = 0.0f;
    float accf[16];
#pragma unroll
    for (int j = 0; j < 16; ++j) accf[j] = 0.0f;

    for (int c0 = 0; c0 < NN; c0 += 64) {
        // ---- async staging: adj 32x64, H 64x128, s2 64
        {
            const float* ap = adj + (size_t)(r0 + row) * NN + c0 + g * 8;
            async_ld16(&Pl[row * 64 + g * 8],     ap);
            async_ld16(&Pl[row * 64 + g * 8 + 4], ap + 4);
#pragma unroll
            for (int i = 0; i < 8; ++i) {
                int idx = i * 256 + tid;        // 0..2047
                int c  = idx >> 5;
                int f4 = idx & 31;
                async_ld16(&Hc[c * FF + f4 * 4],
                           Hb + (size_t)(c0 + c) * FF + f4 * 4);
            }
            if (tid < 16)
                async_ld16(&s2c[tid * 4], s2 + c0 + tid * 4);
        }
        async_wait0();
        __syncthreads();

        // ---- logits + chunk max over my 8 columns
        float e[8];
        float chmax = -3.0e38f;
#pragma unroll
        for (int j = 0; j < 8; ++j) {
            int c = g * 8 + j;
            float ee = lrelu(s1v + s2c[c]);
            if (Pl[row * 64 + c] <= 0.0f) ee = -1.0e12f;
            e[j] = ee;
            chmax = fmaxf(chmax, ee);
        }
#pragma unroll
        for (int off = 1; off < 8; off <<= 1)
            chmax = fmaxf(chmax, __shfl_xor(chmax, off, 32));

        float newmax = fmaxf(mrun, chmax);
        float scale  = __expf(mrun - newmax);

        float psum = 0.0f;
#pragma unroll
        for (int j = 0; j < 8; ++j) {
            float p = __expf(e[j] - newmax);
            Pl[row * 64 + g * 8 + j] = p;   // overwrite own adj slots with P
            psum += p;
        }
#pragma unroll
        for (int off = 1; off < 8; off <<= 1)
            psum += __shfl_xor(psum, off, 32);

        lrun = lrun * scale + psum;
        mrun = newmax;
#pragma unroll
        for (int j = 0; j < 16; ++j) accf[j] *= scale;

        __syncthreads();

        // ---- acc[f0..f0+15] += sum_c P[row][c] * H[c][f]
        for (int c = 0; c < 64; ++c) {
            float pv = Pl[row * 64 + c];
            const float* hp = &Hc[c * FF + f0];
#pragma unroll
            for (int j = 0; j < 16; ++j)
                accf[j] = fmaf(pv, hp[j], accf[j]);
        }
        __syncthreads();
    }

    float inv = 1.0f / lrun;
    float* op = Out + (size_t)(r0 + row) * FF + f0;
#pragma unroll
    for (int j = 0; j < 16; ++j)
        op[j] = lrelu(accf[j] * inv);
}

// ---------------------------------------------------------------------------
// Joint product-of-experts + three predictors (Z=64 -> PH=128 -> Y=5, softmax).
// One thread per row; weight reads are wave-uniform (scalar-cached).
// ---------------------------------------------------------------------------
__device__ __forceinline__ void predict5(const float* z,
    const float* __restrict__ W1, const float* __restrict__ b1,
    const float* __restrict__ W2, const float* __restrict__ b2,
    float* __restrict__ y5)
{
    float o[5];
#pragma unroll
    for (int q = 0; q < 5; ++q) o[q] = b2[q];
    for (int j = 0; j < PHH; ++j) {
        float h = b1[j];
#pragma unroll 8
        for (int f = 0; f < ZZ; ++f)
            h = fmaf(z[f], W1[f * PHH + j], h);
        h = lrelu(h);
#pragma unroll
        for (int q = 0; q < 5; ++q)
            o[q] = fmaf(h, W2[j * 5 + q], o[q]);
    }
    float mx = o[0];
#pragma unroll
    for (int q = 1; q < 5; ++q) mx = fmaxf(mx, o[q]);
    float s = 0.0f;
#pragma unroll
    for (int q = 0; q < 5; ++q) { o[q] = __expf(o[q] - mx); s += o[q]; }
    float inv = 1.0f / s;
#pragma unroll
    for (int q = 0; q < 5; ++q) y5[q] = o[q] * inv;
}

__global__ __launch_bounds__(256) void joint_predict(
    const float* __restrict__ out0, const float* __restrict__ out1,
    const float* __restrict__ mask,
    const float* __restrict__ jpW1, const float* __restrict__ jpb1,
    const float* __restrict__ jpW2, const float* __restrict__ jpb2,
    const float* __restrict__ s0W1, const float* __restrict__ s0b1,
    const float* __restrict__ s0W2, const float* __restrict__ s0b2,
    const float* __restrict__ s1W1, const float* __restrict__ s1b1,
    const float* __restrict__ s1W2, const float* __restrict__ s1b2,
    float* __restrict__ Y)
{
    int i = blockIdx.x * 256 + threadIdx.x;
    float z[ZZ];
    const float mk0 = mask[i * 2 + 0];
    const float mk1 = mask[i * 2 + 1];
    const float* r0 = out0 + (size_t)i * FF;
    const float* r1 = out1 + (size_t)i * FF;

#pragma unroll 4
    for (int f = 0; f < ZZ; ++f) {
        float mu0 = r0[f], lv0 = r0[ZZ + f];
        float mu1 = r1[f], lv1 = r1[ZZ + f];
        float p0 = 1.0f / (__expf(lv0) + 1e-8f);
        float p1 = 1.0f / (__expf(lv1) + 1e-8f);
        float tmp = 1.0f + mk0 * p0 + mk1 * p1;
        float jv  = 1.0f / (tmp + 1e-8f);
        z[f] = jv * (mk0 * p0 * mu0 + mk1 * p1 * mu1);
    }
    predict5(z, jpW1, jpb1, jpW2, jpb2, Y + (size_t)i * 15 + 0);

#pragma unroll 4
    for (int f = 0; f < ZZ; ++f) z[f] = r0[f];
    predict5(z, s0W1, s0b1, s0W2, s0b2, Y + (size_t)i * 15 + 5);

#pragma unroll 4
    for (int f = 0; f < ZZ; ++f) z[f] = r1[f];
    predict5(z, s1W1, s1b1, s1W2, s1b2, Y + (size_t)i * 15 + 10);
}

// ---------------------------------------------------------------------------
extern "C" void kernel_launch(void* const* d_in, const int* in_sizes, int n_in,
                              void* d_out, int out_size, void* d_ws, size_t ws_size,
                              hipStream_t stream)
{
    (void)in_sizes; (void)n_in; (void)out_size; (void)ws_size;

    // setup_inputs() dict order: 12 per modality, then mask, jp*
    const float* x[2]    = { (const float*)d_in[0],  (const float*)d_in[12] };
    const float* adj[2]  = { (const float*)d_in[1],  (const float*)d_in[13] };
    const float* g1W[2]  = { (const float*)d_in[2],  (const float*)d_in[14] };
    const float* g1b[2]  = { (const float*)d_in[3],  (const float*)d_in[15] };
    const float* g2W[2]  = { (const float*)d_in[4],  (const float*)d_in[16] };
    const float* g2b[2]  = { (const float*)d_in[5],  (const float*)d_in[17] };
    const float* gatW[2] = { (const float*)d_in[6],  (const float*)d_in[18] };
    const float* gata[2] = { (const float*)d_in[7],  (const float*)d_in[19] };
    const float* spW1[2] = { (const float*)d_in[8],  (const float*)d_in[20] };
    const float* spb1[2] = { (const float*)d_in[9],  (const float*)d_in[21] };
    const float* spW2[2] = { (const float*)d_in[10], (const float*)d_in[22] };
    const float* spb2[2] = { (const float*)d_in[11], (const float*)d_in[23] };
    const float* mask = (const float*)d_in[24];
    const float* jpW1 = (const float*)d_in[25];
    const float* jpb1 = (const float*)d_in[26];
    const float* jpW2 = (const float*)d_in[27];
    const float* jpb2 = (const float*)d_in[28];

    // workspace layout (floats), ~14.7 MB total
    float* P    = (float*)d_ws;            // 4096*256
    float* Q    = P    + (size_t)NN * HH;  // 4096*256
    float* Hb   = Q    + (size_t)NN * HH;  // 4096*128
    float* s1   = Hb   + (size_t)NN * FF;  // 4096
    float* s2   = s1   + NN;               // 4096
    float* outm[2];
    outm[0]     = s2   + NN;               // 4096*128 (mu0 || lv0)
    outm[1]     = outm[0] + (size_t)NN * FF;

    dim3 blk(256);
    for (int m = 0; m < 2; ++m) {
        // S = x @ W1                       [4096,400]x[400,256]
        gemm_wmma_f32<<<dim3(HH / 128, NN / 128), blk, 0, stream>>>(
            x[m], g1W[m], nullptr, P, NN, HH, DD, 0, 0);
        // X1 = lrelu(adj @ S + b1)         [4096,4096]x[4096,256]
        gemm_wmma_f32<<<dim3(HH / 128, NN / 128), blk, 0, stream>>>(
            adj[m], P, g1b[m], Q, NN, HH, NN, 1, 1);
        // T = X1 @ W2                      [4096,256]x[256,256]
        gemm_wmma_f32<<<dim3(HH / 128, NN / 128), blk, 0, stream>>>(
            Q, g2W[m], nullptr, P, NN, HH, HH, 0, 0);
        // X2 = lrelu(adj @ T + b2)
        gemm_wmma_f32<<<dim3(HH / 128, NN / 128), blk, 0, stream>>>(
            adj[m], P, g2b[m], Q, NN, HH, NN, 1, 1);
        // H = X2 @ Wg                      [4096,256]x[256,128]
        gemm_wmma_f32<<<dim3(FF / 128, NN / 128), blk, 0, stream>>>(
            Q, gatW[m], nullptr, Hb, NN, FF, HH, 0, 0);
        // s1/s2 row dots with attention vector
        rowdots<<<dim3(NN / 256), blk, 0, stream>>>(Hb, gata[m], s1, s2);
        // masked softmax attention + att @ H -> lrelu -> (mu || logvar)
        gat_attention<<<dim3(NN / 32), blk, 0, stream>>>(adj[m], Hb, s1, s2, outm[m]);
    }

    joint_predict<<<dim3(NN / 256), blk, 0, stream>>>(
        outm[0], outm[1], mask,
        jpW1, jpb1, jpW2, jpb2,
        spW1[0], spb1[0], spW2[0], spb2[0],
        spW1[1], spb1[1], spW2[1], spb2[1],
        (float*)d_out);
}